// DAULoss_8272107012511
// MI455X (gfx1250) — compile-verified
//
#include <hip/hip_runtime.h>

typedef float v2f __attribute__((ext_vector_type(2)));
typedef float v8f __attribute__((ext_vector_type(8)));

static constexpr int NROWS = 8192;
static constexpr int NDIM  = 128;

__device__ __forceinline__ float waveReduce(float v) {
#pragma unroll
  for (int off = 16; off > 0; off >>= 1) v += __shfl_xor(v, off, 32);
  return v;
}

// acc[0] = alignment sum, acc[1] = sum_e for X, acc[2] = sum_e for Y
__global__ void dau_init(double* acc) {
  if (threadIdx.x < 3) acc[threadIdx.x] = 0.0;
}

// One wave (32 lanes x float4) per row: normalize, write X-hat / Y-hat,
// per-row sum-of-squares of the normalized rows (matches reference's sn),
// and the alignment partial sum.
__global__ void dau_normalize_align(const float* __restrict__ X,
                                    const float* __restrict__ Y,
                                    float* __restrict__ Xn,
                                    float* __restrict__ Yn,
                                    float* __restrict__ snX,
                                    float* __restrict__ snY,
                                    double* __restrict__ acc) {
  const int lane = threadIdx.x & 31;
  const int wave = threadIdx.x >> 5;
  const int row  = blockIdx.x * 8 + wave;
  const size_t base = (size_t)row * NDIM + lane * 4;

  float4 x = *(const float4*)(X + base);
  float4 y = *(const float4*)(Y + base);

  float ssx = waveReduce(x.x * x.x + x.y * x.y + x.z * x.z + x.w * x.w);
  float ssy = waveReduce(y.x * y.x + y.y * y.y + y.z * y.z + y.w * y.w);
  float ix = 1.0f / fmaxf(sqrtf(ssx), 1e-12f);
  float iy = 1.0f / fmaxf(sqrtf(ssy), 1e-12f);

  float4 xh = make_float4(x.x * ix, x.y * ix, x.z * ix, x.w * ix);
  float4 yh = make_float4(y.x * iy, y.y * iy, y.z * iy, y.w * iy);
  *(float4*)(Xn + base) = xh;
  *(float4*)(Yn + base) = yh;

  float sx = waveReduce(xh.x * xh.x + xh.y * xh.y + xh.z * xh.z + xh.w * xh.w);
  float sy = waveReduce(yh.x * yh.x + yh.y * yh.y + yh.z * yh.z + yh.w * yh.w);

  float d0 = xh.x - yh.x, d1 = xh.y - yh.y, d2 = xh.z - yh.z, d3 = xh.w - yh.w;
  float ad = waveReduce(d0 * d0 + d1 * d1 + d2 * d2 + d3 * d3);

  if (lane == 0) {
    snX[row] = sx;
    snY[row] = sy;
    atomicAdd(acc, (double)ad);
  }
}

// Fused Gram (X-hat * X-hat^T via V_WMMA_F32_16X16X4_F32) + exp + reduce.
// Wave tile: 64 (i) x 32 (j).  Block: 8 waves -> 128 x 128.  Grid: 64 x 64.
__global__ __launch_bounds__(256) void dau_gram_uniform(
    const float* __restrict__ Xn, const float* __restrict__ sn,
    double* __restrict__ accSum) {
  const int lane = threadIdx.x & 31;
  const int wave = threadIdx.x >> 5;
  const int wi = wave & 1;   // 0..1 : i-half of 128-row block
  const int wj = wave >> 1;  // 0..3 : j-quarter of 128-col block
  const int ibase = blockIdx.x * 128 + wi * 64;
  const int jbase = blockIdx.y * 128 + wj * 32;

  const int r  = lane & 15;          // row within 16-row fragment
  const int kh = (lane >> 4) << 1;   // lanes 16-31 hold k+2,k+3

  v8f acc[4][2];
  const v8f vzero = {};
#pragma unroll
  for (int a = 0; a < 4; ++a)
#pragma unroll
    for (int b = 0; b < 2; ++b) acc[a][b] = vzero;

  for (int k0 = 0; k0 < NDIM; k0 += 4) {
    v2f af[4], bf[2];
#pragma unroll
    for (int t = 0; t < 4; ++t)
      af[t] = *(const v2f*)(Xn + (size_t)(ibase + t * 16 + r) * NDIM + k0 + kh);
#pragma unroll
    for (int t = 0; t < 2; ++t)
      bf[t] = *(const v2f*)(Xn + (size_t)(jbase + t * 16 + r) * NDIM + k0 + kh);
#pragma unroll
    for (int ti = 0; ti < 4; ++ti)
#pragma unroll
      for (int tj = 0; tj < 2; ++tj)
        acc[ti][tj] = __builtin_amdgcn_wmma_f32_16x16x4_f32(
            false, af[ti], false, bf[tj], (short)0, acc[ti][tj], false, false);
  }

  // Epilogue: C layout -> VGPR v, lanes 0-15: M=v, N=lane; lanes 16-31: M=8+v.
  const int mo = (lane >> 4) * 8;
  const int jc = lane & 15;
  const float snj0 = sn[jbase + jc];
  const float snj1 = sn[jbase + 16 + jc];

  float lsum = 0.0f;
#pragma unroll
  for (int ti = 0; ti < 4; ++ti) {
#pragma unroll
    for (int v = 0; v < 8; ++v) {
      const int i = ibase + ti * 16 + mo + v;
      const float sni = sn[i];
      {
        const int j = jbase + jc;
        const float d2 = sni + snj0 - 2.0f * acc[ti][0][v];
        const float e = expf(-2.0f * d2);
        if (i != j) lsum += e;
      }
      {
        const int j = jbase + 16 + jc;
        const float d2 = sni + snj1 - 2.0f * acc[ti][1][v];
        const float e = expf(-2.0f * d2);
        if (i != j) lsum += e;
      }
    }
  }
  lsum = waveReduce(lsum);
  if (lane == 0) atomicAdd(accSum, (double)lsum);
}

__global__ void dau_finalize(const double* __restrict__ acc,
                             float* __restrict__ out) {
  const double n = (double)NROWS;
  const double align = acc[0] / n;
  const double denom = n * (n - 1.0);
  const double ux = log(acc[1] / denom);
  const double uy = log(acc[2] / denom);
  out[0] = (float)(align + 0.5 * (ux + uy));
}

extern "C" void kernel_launch(void* const* d_in, const int* in_sizes, int n_in,
                              void* d_out, int out_size, void* d_ws,
                              size_t ws_size, hipStream_t stream) {
  const float* X = (const float*)d_in[0];
  const float* Y = (const float*)d_in[1];
  float* out = (float*)d_out;

  float* Xn  = (float*)d_ws;
  float* Yn  = Xn + (size_t)NROWS * NDIM;
  float* snX = Yn + (size_t)NROWS * NDIM;
  float* snY = snX + NROWS;
  double* acc = (double*)(snY + NROWS);  // 3 doubles, 8-byte aligned offset

  dau_init<<<1, 32, 0, stream>>>(acc);
  dau_normalize_align<<<NROWS / 8, 256, 0, stream>>>(X, Y, Xn, Yn, snX, snY,
                                                     acc);
  dim3 grid(NROWS / 128, NROWS / 128);
  dau_gram_uniform<<<grid, 256, 0, stream>>>(Xn, snX, acc + 1);
  dau_gram_uniform<<<grid, 256, 0, stream>>>(Yn, snY, acc + 2);
  dau_finalize<<<1, 1, 0, stream>>>(acc, out);
}